// GELU226_23648089932062
// MI455X (gfx1250) — compile-verified
//
#include <hip/hip_runtime.h>
#include <cmath>

// Problem constants (match reference: B=4, T=4096, D=2048, N=32)
#define D_DIM    2048
#define ROWS     16384            // B*T
#define N_BUF    32
#define ROW_TILE 64
#define N_TILES  (ROWS / ROW_TILE)   // 256
#define TOTAL4   (ROWS * (D_DIM / 4))// 8388608 float4 elements

typedef __attribute__((ext_vector_type(2))) float v2f;
typedef __attribute__((ext_vector_type(8))) float v8f;

__device__ __forceinline__ float gelu_tanh(float x) {
    const float c = 0.7978845608028654f;  // sqrt(2/pi)
    float inner = c * (x + 0.044715f * x * x * x);
    return 0.5f * x * (1.0f + tanhf(inner));
}

// ---------------------------------------------------------------------------
// Kernel 1: y = gelu(x) column partial sums. grid = (2, N_TILES), block = 256.
// Each block: 64 rows x 1024 columns, float4 per thread per row. 128 MB read.
// ---------------------------------------------------------------------------
__global__ void k_gelu_colsum(const float* __restrict__ x,
                              float* __restrict__ partial) {
    const int colBase = blockIdx.x * 1024 + threadIdx.x * 4;
    const int row0    = blockIdx.y * ROW_TILE;
    const float4* xv  = reinterpret_cast<const float4*>(x);

    int idx4 = (row0 * D_DIM + colBase) >> 2;   // float4 index, < 2^23
    float4 acc = make_float4(0.f, 0.f, 0.f, 0.f);

    for (int r = 0; r < ROW_TILE; ++r) {
        // Prefetch 8 rows ahead (speculative; clamped to stay in-bounds).
        int pf = idx4 + 8 * (D_DIM / 4);
        if (pf >= TOTAL4) pf = TOTAL4 - 1;
        __builtin_prefetch(&xv[pf], 0, 3);      // -> global_prefetch_b8

        float4 v = xv[idx4];
        acc.x += gelu_tanh(v.x);
        acc.y += gelu_tanh(v.y);
        acc.z += gelu_tanh(v.z);
        acc.w += gelu_tanh(v.w);
        idx4 += D_DIM / 4;
    }
    float4* pv = reinterpret_cast<float4*>(partial);
    pv[(blockIdx.y * D_DIM + colBase) >> 2] = acc;
}

// ---------------------------------------------------------------------------
// Kernel 2: deterministic reduce of tile partials -> m_curr[D]. grid = 8x256.
// ---------------------------------------------------------------------------
__global__ void k_reduce_mean(const float* __restrict__ partial,
                              float* __restrict__ m_curr) {
    const int d = blockIdx.x * 256 + threadIdx.x;
    float s = 0.f;
    for (int t = 0; t < N_TILES; ++t) s += partial[t * D_DIM + d];
    m_curr[d] = s * (1.0f / (float)ROWS);
}

// ---------------------------------------------------------------------------
// Kernel 3 (single block, 256 threads):
//   - ||m||^2 via block reduction
//   - wave 0: WMMA f32 16x16x4 over K=2048:
//       cd += A(buf tile 16x4) x B(m broadcast 4x16)   -> dots
//       cg += A x A (Gram tile; A/B lane layouts match) -> diag = row norms^2
//   - serial argmax (first-max tie-break like jnp.argmax), buf std, scalars
// ---------------------------------------------------------------------------
__global__ void k_stats(const float* __restrict__ m_curr_g,
                        const float* __restrict__ buf,
                        const unsigned char* __restrict__ mask,   // numpy bool: 1 byte
                        const float* __restrict__ log_alpha,
                        const float* __restrict__ log_tau,
                        float* __restrict__ nearest,
                        float* __restrict__ invstd,
                        float* __restrict__ scalars) {
    __shared__ float m[D_DIM];
    __shared__ float red[256];
    __shared__ float dots[N_BUF];
    __shared__ float norm2[N_BUF];
    __shared__ int   best;
    const int tid = threadIdx.x;

    for (int d = tid; d < D_DIM; d += 256) m[d] = m_curr_g[d];
    __syncthreads();

    // ||m||^2
    float s = 0.f;
    for (int d = tid; d < D_DIM; d += 256) s += m[d] * m[d];
    red[tid] = s;
    __syncthreads();
    for (int off = 128; off > 0; off >>= 1) {
        if (tid < off) red[tid] += red[tid + off];
        __syncthreads();
    }

    // WMMA sims + norms on wave 0 (EXEC all-1s inside the uniform wave branch)
    if (tid < 32) {
        const int lane = tid;
        const int grp2 = (lane >> 4) << 1;  // lanes 0-15 hold K+0/1, 16-31 K+2/3
        const int mrow = lane & 15;
        for (int rt = 0; rt < 2; ++rt) {
            const float* arow = buf + (rt * 16 + mrow) * D_DIM + grp2;
            v8f cd = {};
            v8f cg = {};
            for (int k0 = 0; k0 < D_DIM; k0 += 4) {
                v2f a; a.x = arow[k0];        a.y = arow[k0 + 1];
                v2f b; b.x = m[k0 + grp2];    b.y = m[k0 + grp2 + 1];  // broadcast over N
                cd = __builtin_amdgcn_wmma_f32_16x16x4_f32(
                         false, a, false, b, (short)0, cd, false, false);
                cg = __builtin_amdgcn_wmma_f32_16x16x4_f32(
                         false, a, false, a, (short)0, cg, false, false);
            }
            // D layout: VGPR v, lanes 0-15 -> M=v (N=lane), lanes 16-31 -> M=v+8
            if (lane == 0) {
#pragma unroll
                for (int v = 0; v < 8; ++v) dots[rt * 16 + v] = cd[v];
            }
            if (lane == 16) {
#pragma unroll
                for (int v = 0; v < 8; ++v) dots[rt * 16 + 8 + v] = cd[v];
            }
            if (lane < 8)   norm2[rt * 16 + lane] = cg[lane];              // D[M][M], M<8
            if (lane >= 24) norm2[rt * 16 + 8 + (lane - 24)] = cg[lane - 24]; // M>=8
        }
    }
    __syncthreads();

    if (tid == 0) {
        float mn = fmaxf(sqrtf(red[0]), 1e-12f);   // F.normalize eps
        float bestv = -2.0f;
        int bi = 0;
        for (int i = 0; i < N_BUF; ++i) {
            float bn  = fmaxf(sqrtf(norm2[i]), 1e-12f);
            float sim = mask[i] ? (dots[i] / (bn * mn)) : -1.0f;
            if (sim > bestv) { bestv = sim; bi = i; }  // first-max tie-break
        }
        best = bi;
        scalars[0] = log1pf(expf(log_alpha[0]));   // softplus(log_alpha)
        scalars[1] = expf(log_tau[0]);             // tau
    }
    __syncthreads();

    const int bi = best;
    for (int d = tid; d < D_DIM; d += 256) {
        float sw = 0.f, s1 = 0.f, s2 = 0.f;
        for (int i = 0; i < N_BUF; ++i) {
            float w = mask[i] ? 1.0f : 0.0f;
            float v = buf[i * D_DIM + d];
            sw += w; s1 += w * v; s2 += w * v * v;
        }
        float mu  = s1 / sw;
        float var = fmaxf((s2 - sw * mu * mu) / (sw - 1.0f), 0.0f);
        float sd  = fmaxf(sqrtf(var), 1e-4f);
        invstd[d]  = 1.0f / (sd + 1e-5f);
        nearest[d] = buf[bi * D_DIM + d];
    }
}

// ---------------------------------------------------------------------------
// Kernel 4: out = y * gate. Recompute gelu (x is L2-resident: 128MB < 192MB).
// grid = 1024 blocks x 256 threads x 32 float4 each = 32M elements.
// ---------------------------------------------------------------------------
__global__ void k_apply(const float* __restrict__ x,
                        const float* __restrict__ nearest,
                        const float* __restrict__ invstd,
                        const float* __restrict__ scalars,
                        float* __restrict__ out) {
    __shared__ float s_near[D_DIM];
    __shared__ float s_istd[D_DIM];
    for (int d = threadIdx.x; d < D_DIM; d += 256) {
        s_near[d] = nearest[d];
        s_istd[d] = invstd[d];
    }
    __syncthreads();

    const float alpha = scalars[0];
    const float tau   = scalars[1];

    const float4* xv = reinterpret_cast<const float4*>(x);
    float4*       ov = reinterpret_cast<float4*>(out);

    const int PER  = 32;
    const int base = blockIdx.x * (256 * PER) + threadIdx.x;

    for (int i = 0; i < PER; ++i) {
        int id4 = base + i * 256;
        int pf  = id4 + 2048;
        if (pf >= TOTAL4) pf = TOTAL4 - 1;
        __builtin_prefetch(&xv[pf], 0, 3);      // -> global_prefetch_b8

        float4 v = xv[id4];
        const int col = (id4 & ((D_DIM / 4) - 1)) << 2;   // (id4*4) % D

        float4 r;
        {
            float y = gelu_tanh(v.x);
            float dev = (y - s_near[col + 0]) * s_istd[col + 0];
            float fam = expf(-tau * dev * dev);
            r.x = y * fminf(fmaxf(1.0f - alpha * fam, 0.05f), 1.05f);
        }
        {
            float y = gelu_tanh(v.y);
            float dev = (y - s_near[col + 1]) * s_istd[col + 1];
            float fam = expf(-tau * dev * dev);
            r.y = y * fminf(fmaxf(1.0f - alpha * fam, 0.05f), 1.05f);
        }
        {
            float y = gelu_tanh(v.z);
            float dev = (y - s_near[col + 2]) * s_istd[col + 2];
            float fam = expf(-tau * dev * dev);
            r.z = y * fminf(fmaxf(1.0f - alpha * fam, 0.05f), 1.05f);
        }
        {
            float y = gelu_tanh(v.w);
            float dev = (y - s_near[col + 3]) * s_istd[col + 3];
            float fam = expf(-tau * dev * dev);
            r.w = y * fminf(fmaxf(1.0f - alpha * fam, 0.05f), 1.05f);
        }
        ov[id4] = r;
    }
}

// ---------------------------------------------------------------------------
// Launch. Inputs (reference order): x[f32 32M], buf[f32 64K], mask[bool 32],
// log_alpha[f32 1], log_tau[f32 1]. Output: f32 32M.
// ws layout (floats): partial[256*2048] | m_curr[2048] | nearest[2048] |
//                     invstd[2048] | scalars[2]   (~2.03 MB)
// ---------------------------------------------------------------------------
extern "C" void kernel_launch(void* const* d_in, const int* in_sizes, int n_in,
                              void* d_out, int out_size, void* d_ws, size_t ws_size,
                              hipStream_t stream) {
    const float*         x    = (const float*)d_in[0];
    const float*         buf  = (const float*)d_in[1];
    const unsigned char* mask = (const unsigned char*)d_in[2];  // numpy bool = 1 byte
    const float*         la   = (const float*)d_in[3];
    const float*         lt   = (const float*)d_in[4];
    float*               out  = (float*)d_out;

    float* ws      = (float*)d_ws;
    float* partial = ws;                                  // 256*2048
    float* m_curr  = partial + N_TILES * D_DIM;           // 2048
    float* nearest = m_curr + D_DIM;                      // 2048
    float* invstd  = nearest + D_DIM;                     // 2048
    float* scalars = invstd + D_DIM;                      // 2

    dim3 g1(2, N_TILES);
    k_gelu_colsum<<<g1, 256, 0, stream>>>(x, partial);
    k_reduce_mean<<<D_DIM / 256, 256, 0, stream>>>(partial, m_curr);
    k_stats<<<1, 256, 0, stream>>>(m_curr, buf, mask, la, lt, nearest, invstd, scalars);
    k_apply<<<1024, 256, 0, stream>>>(x, nearest, invstd, scalars, out);
}